// GCN_72756745994559
// MI455X (gfx1250) — compile-verified
//
#include <hip/hip_runtime.h>

#define HDIM 128
#define SLOPE 0.01f
#define GN_EPS 1e-5f

typedef __attribute__((ext_vector_type(2))) float v2f;
typedef __attribute__((ext_vector_type(8))) float v8f;

// ---------------- utility kernels ----------------

__global__ void zero_kernel(float* p, int n) {
    int i = blockIdx.x * blockDim.x + threadIdx.x;
    if (i < n) p[i] = 0.0f;
}

__global__ void degree_kernel(const int* __restrict__ dst, int E, float* deg) {
    int e = blockIdx.x * blockDim.x + threadIdx.x;
    if (e < E) unsafeAtomicAdd(&deg[dst[e]], 1.0f);
}

__global__ void dis_kernel(const float* __restrict__ deg, float* dis, float* dis2, int N) {
    int i = blockIdx.x * blockDim.x + threadIdx.x;
    if (i < N) {
        float d = deg[i] + 1.0f;       // self loop added
        dis[i]  = rsqrtf(d);           // deg^{-1/2}
        dis2[i] = 1.0f / d;            // (deg^{-1/2})^2
    }
}

// Per-feature sum / sumsq over nodes, optionally fusing LeakyReLU write-back.
// blockDim.x == HDIM; thread t owns feature t; grid-stride over rows.
__global__ void stats_kernel(const float* __restrict__ in, float* out,
                             int N, float* sum, float* sumsq, int applyLeaky) {
    int f = threadIdx.x;
    float s = 0.0f, s2 = 0.0f;
    for (int r = blockIdx.x; r < N; r += gridDim.x) {
        float v = in[(size_t)r * HDIM + f];
        if (applyLeaky) {
            v = (v > 0.0f) ? v : SLOPE * v;
            out[(size_t)r * HDIM + f] = v;
        }
        s  += v;
        s2 += v * v;
    }
    unsafeAtomicAdd(&sum[f], s);
    unsafeAtomicAdd(&sumsq[f], s2);
}

// GraphNorm folded to per-feature affine: y = x*alpha + beta
__global__ void prep_affine(const float* __restrict__ sum, const float* __restrict__ sumsq,
                            const float* __restrict__ w, const float* __restrict__ b,
                            const float* __restrict__ s, float* alpha, float* beta, int N) {
    int f = threadIdx.x;
    float invN = 1.0f / (float)N;
    float m  = sum[f] * invN;
    float e2 = sumsq[f] * invN;
    float sv = s[f];
    float var = e2 - 2.0f * sv * m * m + sv * sv * m * m;  // E[(x - s*m)^2]
    float r = rsqrtf(var + GN_EPS);
    float a = r * w[f];
    alpha[f] = a;
    beta[f]  = b[f] - sv * m * a;
}

// ---------------- WMMA GEMM ----------------
// Out[N x NOUT] = affine(A[N x 128]) @ W[128 x NOUT]
// One block = MTILES 16-row tiles (80 rows); wave w owns output columns [16w,16w+16).
// Per k-step: one B fragment feeds MTILES WMMAs (5x less B cache traffic).
template <int NOUT, int MTILES>
__global__ void gemm_norm_wmma(const float* __restrict__ A,
                               const float* __restrict__ alpha,
                               const float* __restrict__ beta,
                               const float* __restrict__ W,   // row-major [128][NOUT]
                               float* __restrict__ Out,
                               int N) {
    constexpr int K = HDIM;
    constexpr int LSTR = 132;                  // padded LDS row stride (floats)
    __shared__ float As[MTILES * 16 * LSTR];
    const int row0 = blockIdx.x * (MTILES * 16);

    // Cooperative float4 stage of the A tile with the GraphNorm affine fused.
    const int nf4 = MTILES * 16 * (K / 4);
    for (int idx = threadIdx.x; idx < nf4; idx += blockDim.x) {
        int r = idx >> 5;                      // K/4 == 32 float4 per row
        int q = idx & 31;
        int row = row0 + r;
        if (row >= N) row = N - 1;             // harmless clamp (N%16==0)
        float4 v  = ((const float4*)(A + (size_t)row * K))[q];
        float4 al = ((const float4*)alpha)[q];
        float4 be = ((const float4*)beta)[q];
        float4 o;
        o.x = v.x * al.x + be.x;
        o.y = v.y * al.y + be.y;
        o.z = v.z * al.z + be.z;
        o.w = v.w * al.w + be.w;
        *((float4*)&As[r * LSTR + q * 4]) = o;
    }
    __syncthreads();

    const int lane = threadIdx.x & 31;
    const int wave = threadIdx.x >> 5;
    const int col0 = wave * 16;
    const int mn   = lane & 15;   // row (A) / column (B,D) within tile
    const int kh   = lane >> 4;   // K half-select per ISA 16x16x4 f32 layout

    v8f acc[MTILES];
#pragma unroll
    for (int t = 0; t < MTILES; ++t) acc[t] = (v8f){};

    for (int k0 = 0; k0 < K; k0 += 4) {
        v2f b;
        b.x = W[(size_t)(k0 + 2 * kh)     * NOUT + col0 + mn];
        b.y = W[(size_t)(k0 + 2 * kh + 1) * NOUT + col0 + mn];
#pragma unroll
        for (int t = 0; t < MTILES; ++t) {
            const float* ap = &As[(t * 16 + mn) * LSTR + k0 + 2 * kh];
            v2f a;
            a.x = ap[0];
            a.y = ap[1];
            acc[t] = __builtin_amdgcn_wmma_f32_16x16x4_f32(
                false, a, false, b, (short)0, acc[t], false, false);
        }
    }

#pragma unroll
    for (int t = 0; t < MTILES; ++t) {
        int rowt = row0 + t * 16 + 8 * kh;     // first of 8 consecutive rows
        if (rowt + 8 <= N) {
            float* op = Out + (size_t)rowt * NOUT + col0 + mn;
#pragma unroll
            for (int vv = 0; vv < 8; ++vv)
                op[(size_t)vv * NOUT] = acc[t][vv];
        } else {
            for (int vv = 0; vv < 8; ++vv) {
                int row = rowt + vv;
                if (row < N) Out[(size_t)row * NOUT + col0 + mn] = acc[t][vv];
            }
        }
    }
}

// agg = h * dis2[node] + bias[f]   (self-loop + bias, initializes agg buffer)
// operates on float4 granularity; F4 = F/4
__global__ void selfloop_init(const float4* __restrict__ h4, const float* __restrict__ dis2,
                              const float4* __restrict__ bias4, float4* agg4,
                              int N, int F4) {
    size_t idx = (size_t)blockIdx.x * blockDim.x + threadIdx.x;
    if (idx < (size_t)N * F4) {
        int node = (int)(idx / F4);
        int f    = (int)(idx % F4);
        float4 v = h4[idx];
        float4 b = bias4[f];
        float  d = dis2[node];
        float4 o;
        o.x = v.x * d + b.x;
        o.y = v.y * d + b.y;
        o.z = v.z * d + b.z;
        o.w = v.w * d + b.w;
        agg4[idx] = o;
    }
}

// One wave per edge, F=128: float4 gather per lane + 4 fp32 atomics
__global__ void edge_agg128(const float* __restrict__ h, const int* __restrict__ src,
                            const int* __restrict__ dst, const float* __restrict__ dis,
                            float* agg, int E) {
    size_t gid = (size_t)blockIdx.x * blockDim.x + threadIdx.x;
    int e = (int)(gid >> 5);
    int lane = (int)(gid & 31);
    if (e >= E) return;
    int s = src[e];
    int d = dst[e];
    float nrm = dis[s] * dis[d];
    float4 hv = ((const float4*)(h + (size_t)s * 128))[lane];
    float* ad = agg + (size_t)d * 128 + lane * 4;
    unsafeAtomicAdd(&ad[0], hv.x * nrm);
    unsafeAtomicAdd(&ad[1], hv.y * nrm);
    unsafeAtomicAdd(&ad[2], hv.z * nrm);
    unsafeAtomicAdd(&ad[3], hv.w * nrm);
}

// One wave per edge, F=32: one float per lane
__global__ void edge_agg32(const float* __restrict__ h, const int* __restrict__ src,
                           const int* __restrict__ dst, const float* __restrict__ dis,
                           float* agg, int E) {
    size_t gid = (size_t)blockIdx.x * blockDim.x + threadIdx.x;
    int e = (int)(gid >> 5);
    int lane = (int)(gid & 31);
    if (e >= E) return;
    int s = src[e];
    int d = dst[e];
    float nrm = dis[s] * dis[d];
    unsafeAtomicAdd(&agg[(size_t)d * 32 + lane], h[(size_t)s * 32 + lane] * nrm);
}

// ---------------- launch ----------------

extern "C" void kernel_launch(void* const* d_in, const int* in_sizes, int n_in,
                              void* d_out, int out_size, void* d_ws, size_t ws_size,
                              hipStream_t stream) {
    const int N = in_sizes[0] / HDIM;
    const int E = in_sizes[1] / 2;
    const int O = 32;
    constexpr int MT = 5;                      // 3125 16-row tiles = 5 * 625

    const float* x    = (const float*)d_in[0];
    const int*   ei   = (const int*)d_in[1];
    const float* gn_w = (const float*)d_in[2];
    const float* gn_b = (const float*)d_in[3];
    const float* gn_s = (const float*)d_in[4];
    const float* Wh   = (const float*)d_in[5];
    const float* bh   = (const float*)d_in[6];
    const float* Wo   = (const float*)d_in[7];
    const float* bo   = (const float*)d_in[8];
    float* out = (float*)d_out;

    const int* src = ei;
    const int* dst = ei + E;

    float* w = (float*)d_ws;
    float* deg   = w; w += N;
    float* dis   = w; w += N;
    float* dis2  = w; w += N;
    float* sum   = w; w += HDIM;
    float* sumsq = w; w += HDIM;
    float* alpha = w; w += HDIM;
    float* beta  = w; w += HDIM;
    float* bufA  = w; w += (size_t)N * HDIM;   // GEMM out (h)
    float* bufB  = w; w += (size_t)N * HDIM;   // agg / activated features

    // Degree + symmetric normalization factors
    zero_kernel<<<(N + 255) / 256, 256, 0, stream>>>(deg, N);
    degree_kernel<<<(E + 255) / 256, 256, 0, stream>>>(dst, E, deg);
    dis_kernel<<<(N + 255) / 256, 256, 0, stream>>>(deg, dis, dis2, N);

    const int STATS_BLOCKS = 512;
    const int gemm_blocks = (N + MT * 16 - 1) / (MT * 16);
    const int eblocks = (int)(((size_t)E * 32 + 255) / 256);

    for (int i = 0; i < 4; ++i) {
        const float* cur = (i == 0) ? x : bufB;
        zero_kernel<<<1, 256, 0, stream>>>(sum, 2 * HDIM);   // sum & sumsq contiguous
        stats_kernel<<<STATS_BLOCKS, HDIM, 0, stream>>>(cur, bufB, N, sum, sumsq, i > 0 ? 1 : 0);
        prep_affine<<<1, HDIM, 0, stream>>>(sum, sumsq, gn_w + i * HDIM, gn_b + i * HDIM,
                                            gn_s + i * HDIM, alpha, beta, N);
        gemm_norm_wmma<HDIM, MT><<<gemm_blocks, 256, 0, stream>>>(
            cur, alpha, beta, Wh + (size_t)i * HDIM * HDIM, bufA, N);
        selfloop_init<<<(int)(((size_t)N * (HDIM / 4) + 255) / 256), 256, 0, stream>>>(
            (const float4*)bufA, dis2, (const float4*)(bh + i * HDIM), (float4*)bufB,
            N, HDIM / 4);
        edge_agg128<<<eblocks, 256, 0, stream>>>(bufA, src, dst, dis, bufB, E);
    }

    // Final block: norm -> conv(W_out, 128x32), no activation
    zero_kernel<<<1, 256, 0, stream>>>(sum, 2 * HDIM);
    stats_kernel<<<STATS_BLOCKS, HDIM, 0, stream>>>(bufB, bufB, N, sum, sumsq, 1);
    prep_affine<<<1, HDIM, 0, stream>>>(sum, sumsq, gn_w + 4 * HDIM, gn_b + 4 * HDIM,
                                        gn_s + 4 * HDIM, alpha, beta, N);
    gemm_norm_wmma<32, MT><<<gemm_blocks, 64, 0, stream>>>(bufB, alpha, beta, Wo, bufA, N);
    selfloop_init<<<(int)(((size_t)N * (O / 4) + 255) / 256), 256, 0, stream>>>(
        (const float4*)bufA, dis2, (const float4*)bo, (float4*)out, N, O / 4);
    edge_agg32<<<eblocks, 256, 0, stream>>>(bufA, src, dst, dis, out, E);
}